// ExactOnlineLSTM_51316269253252
// MI455X (gfx1250) — compile-verified
//
#include <hip/hip_runtime.h>

typedef float v2f __attribute__((ext_vector_type(2)));
typedef float v8f __attribute__((ext_vector_type(8)));

#define NUM_EMB 155
#define T_LEN   512
#define GSTRIDE 68   // 64 gate cols (4 blocks x 16) + 4 pad -> stride%64banks = 4

__device__ __forceinline__ float fexp2(float x) {
#if __has_builtin(__builtin_amdgcn_exp2f)
  return __builtin_amdgcn_exp2f(x);
#else
  return exp2f(x);
#endif
}
__device__ __forceinline__ float frcp(float x) {
#if __has_builtin(__builtin_amdgcn_rcpf)
  return __builtin_amdgcn_rcpf(x);
#else
  return 1.0f / x;
#endif
}
__device__ __forceinline__ float ftanh(float x) {
#if __has_builtin(__builtin_amdgcn_tanhf)
  return __builtin_amdgcn_tanhf(x);            // gfx1250 v_tanh_f32
#else
  float e = fexp2(x * 2.8853900817779268f);    // exp(2x)
  return 1.0f - 2.0f * frcp(e + 1.0f);
#endif
}
__device__ __forceinline__ float fsig(float x) {
  return 0.5f + 0.5f * ftanh(0.5f * x);
}
__device__ __forceinline__ float permx16(float v, int addr) {
  // half-wave swap: lane l gets value from lane l^16 (full-EXEC, convergent)
  return __int_as_float(__builtin_amdgcn_ds_bpermute(addr, __float_as_int(v)));
}
__device__ __forceinline__ v8f wmma4(v2f a, v2f b, v8f c) {
  // D(16x16 f32) = A(16x4 f32) x B(4x16 f32) + C
  return __builtin_amdgcn_wmma_f32_16x16x4_f32(false, a, false, b, (short)0, c,
                                               false, false);
}

__global__ __launch_bounds__(256)
void ExactOnlineLSTM_kernel(const int* __restrict__ x,
                            const float* __restrict__ W_emb,   // [155,8]
                            const float* __restrict__ W_ih,    // [40,8]
                            const float* __restrict__ W_hh,    // [40,10]
                            const float* __restrict__ b_ih,    // [40]
                            const float* __restrict__ b_hh,    // [40]
                            const float* __restrict__ W_head,  // [32,10]
                            const float* __restrict__ b_head,  // [32]
                            float* __restrict__ out)           // [B,32]
{
  __shared__ float Gs[NUM_EMB * GSTRIDE];

  const int tid = threadIdx.x;

  // ---- Build fused token->gate table: G[e][blk*16+j] = W_emb[e]·W_ih[blk*10+j] + bias
  for (int i = tid; i < NUM_EMB * 64; i += 256) {
    int e = i >> 6, col = i & 63, blk = col >> 4, j = col & 15;
    float v = 0.0f;
    if (j < 10) {
      int g = blk * 10 + j;
      float s = b_ih[g] + b_hh[g];
#pragma unroll
      for (int k = 0; k < 8; ++k) s += W_emb[e * 8 + k] * W_ih[g * 8 + k];
      v = s;
    }
    Gs[e * GSTRIDE + col] = v;
  }
  __syncthreads();

  const int  lane  = tid & 31;
  const int  wave  = tid >> 5;
  const int  mrow  = lane & 15;
  const bool lo    = lane < 16;
  const int  kbase = lo ? 0 : 2;      // A/B 16x16x4 f32 layout: high half holds K+2
  const int  paddr = (lane ^ 16) << 2;
  const int  joff  = lo ? 0 : 8;      // hidden-index half this lane owns in D layout
  const int  bglob = blockIdx.x * 128 + wave * 16 + mrow;   // batch column

  // ---- A operands: padded W_hh, gate blocks i/f/g/o each a 16-row M tile, K=12 (3 ksteps)
  v2f a_hh[4][3];
#pragma unroll
  for (int blk = 0; blk < 4; ++blk)
#pragma unroll
    for (int s = 0; s < 3; ++s)
#pragma unroll
      for (int v = 0; v < 2; ++v) {
        int k = 4 * s + kbase + v;
        a_hh[blk][s][v] =
            (mrow < 10 && k < 10) ? W_hh[(blk * 10 + mrow) * 10 + k] : 0.0f;
      }

  float h[8], c[8];
#pragma unroll
  for (int i = 0; i < 8; ++i) { h[i] = 0.0f; c[i] = 0.0f; }

  const int* xrow = x + (size_t)bglob * T_LEN;
  int4 cur = *(const int4*)(xrow);

  for (int t4 = 0; t4 < T_LEN / 4; ++t4) {
    int4 nxt = cur;
    if (t4 + 1 < T_LEN / 4) nxt = *(const int4*)(xrow + 4 * (t4 + 1));  // prefetch
    int ids[4] = {cur.x, cur.y, cur.z, cur.w};

#pragma unroll
    for (int u = 0; u < 4; ++u) {
      int r = ids[u];
      r = r > (NUM_EMB - 1) ? (NUM_EMB - 1) : r;   // torch.clamp(max=154)
      r = r < 0 ? 0 : r;

      // gather x-gate contribution (D layout C operands), bank-conflict-free
      const float* grow = &Gs[r * GSTRIDE + joff];
      v8f d[4];
#pragma unroll
      for (int blk = 0; blk < 4; ++blk) {
        float4 p = *(const float4*)(grow + blk * 16);
        float4 q = *(const float4*)(grow + blk * 16 + 4);
        v8f cc;
        cc[0] = p.x; cc[1] = p.y; cc[2] = p.z; cc[3] = p.w;
        cc[4] = q.x; cc[5] = q.y; cc[6] = q.z; cc[7] = q.w;
        d[blk] = cc;
      }

      // rebuild B (h^T, K x N) from D-layout h via half-wave swaps
      float w0 = permx16(h[0], paddr), w1 = permx16(h[1], paddr);
      float w2 = permx16(h[2], paddr), w3 = permx16(h[3], paddr);
      float w6 = permx16(h[6], paddr), w7 = permx16(h[7], paddr);
      v2f B0 = {lo ? h[0] : w2, lo ? h[1] : w3};   // K=0,1 | 2,3
      v2f B1 = {lo ? h[4] : w6, lo ? h[5] : w7};   // K=4,5 | 6,7
      v2f B2 = {lo ? w0 : 0.0f, lo ? w1 : 0.0f};   // K=8,9 | 10,11(pad)

      // gates^T = W_hh_pad @ h^T + xg  : 12x V_WMMA_F32_16X16X4_F32
#pragma unroll
      for (int blk = 0; blk < 4; ++blk) {
        d[blk] = wmma4(a_hh[blk][0], B0, d[blk]);
        d[blk] = wmma4(a_hh[blk][1], B1, d[blk]);
        d[blk] = wmma4(a_hh[blk][2], B2, d[blk]);
      }

      // elementwise LSTM update — i/f/g/o/c/h all share identical layout
#pragma unroll
      for (int i = 0; i < 8; ++i) {
        float iv = fsig(d[0][i]);
        float fv = fsig(d[1][i]);
        float gv = ftanh(d[2][i]);
        float ov = fsig(d[3][i]);
        float cn = fv * c[i] + iv * gv;
        c[i] = cn;
        h[i] = ov * ftanh(cn);
      }
    }
    cur = nxt;
  }

  // ---- head: out^T = W_head_pad @ h^T + b_head  (2 M-tiles x 3 ksteps)
  v2f a_hd[2][3];
#pragma unroll
  for (int tile = 0; tile < 2; ++tile)
#pragma unroll
    for (int s = 0; s < 3; ++s)
#pragma unroll
      for (int v = 0; v < 2; ++v) {
        int k = 4 * s + kbase + v;
        int m = mrow + 16 * tile;
        a_hd[tile][s][v] = (k < 10) ? W_head[m * 10 + k] : 0.0f;
      }

  float w0 = permx16(h[0], paddr), w1 = permx16(h[1], paddr);
  float w2 = permx16(h[2], paddr), w3 = permx16(h[3], paddr);
  float w6 = permx16(h[6], paddr), w7 = permx16(h[7], paddr);
  v2f B0 = {lo ? h[0] : w2, lo ? h[1] : w3};
  v2f B1 = {lo ? h[4] : w6, lo ? h[5] : w7};
  v2f B2 = {lo ? w0 : 0.0f, lo ? w1 : 0.0f};

  float* orow = out + (size_t)bglob * 32 + joff;
#pragma unroll
  for (int tile = 0; tile < 2; ++tile) {
    const float* bh = b_head + joff + 16 * tile;
    float4 p = *(const float4*)(bh);
    float4 q = *(const float4*)(bh + 4);
    v8f acc;
    acc[0] = p.x; acc[1] = p.y; acc[2] = p.z; acc[3] = p.w;
    acc[4] = q.x; acc[5] = q.y; acc[6] = q.z; acc[7] = q.w;
    acc = wmma4(a_hd[tile][0], B0, acc);
    acc = wmma4(a_hd[tile][1], B1, acc);
    acc = wmma4(a_hd[tile][2], B2, acc);
    float4 s0 = {acc[0], acc[1], acc[2], acc[3]};
    float4 s1 = {acc[4], acc[5], acc[6], acc[7]};
    *(float4*)(orow + 16 * tile)     = s0;
    *(float4*)(orow + 16 * tile + 4) = s1;
  }
}

extern "C" void kernel_launch(void* const* d_in, const int* in_sizes, int n_in,
                              void* d_out, int out_size, void* d_ws, size_t ws_size,
                              hipStream_t stream) {
  const int*   x      = (const int*)d_in[0];
  const float* W_emb  = (const float*)d_in[1];
  const float* W_ih   = (const float*)d_in[2];
  const float* W_hh   = (const float*)d_in[3];
  const float* b_ih   = (const float*)d_in[4];
  const float* b_hh   = (const float*)d_in[5];
  const float* W_head = (const float*)d_in[6];
  const float* b_head = (const float*)d_in[7];
  float* out = (float*)d_out;

  int B = in_sizes[0] / T_LEN;       // 8192
  int grid = B / 128;                // 16 batch/wave * 8 waves/block
  ExactOnlineLSTM_kernel<<<grid, 256, 0, stream>>>(
      x, W_emb, W_ih, W_hh, b_ih, b_hh, W_head, b_head, out);
}